// NGCF_79242146611300
// MI455X (gfx1250) — compile-verified
//
#include <hip/hip_runtime.h>

#define N_USERS 100000
#define N_ITEMS 50000
#define N_NODES (N_USERS + N_ITEMS)   // 150000, divisible by 16
#define N_EDGES 2400000
#define EMB 64
#define LEAKY_ALPHA 0.3f

typedef __attribute__((ext_vector_type(2))) float v2f;
typedef __attribute__((ext_vector_type(8))) float v8f;

// ---------------------------------------------------------------------------
// Kernel 1: e0 = concat(user_emb, item_emb); out = 0.25*e0; s = 0
// ---------------------------------------------------------------------------
__global__ void ngcf_init(const float* __restrict__ user_emb,
                          const float* __restrict__ item_emb,
                          float* __restrict__ e0,
                          float* __restrict__ s,
                          float* __restrict__ out) {
    int i = blockIdx.x * blockDim.x + threadIdx.x;       // float4 index
    const int n4 = N_NODES * EMB / 4;
    if (i >= n4) return;
    const int user4 = N_USERS * EMB / 4;
    float4 v = (i < user4) ? ((const float4*)user_emb)[i]
                           : ((const float4*)item_emb)[i - user4];
    ((float4*)e0)[i] = v;
    ((float4*)out)[i] = make_float4(0.25f * v.x, 0.25f * v.y,
                                    0.25f * v.z, 0.25f * v.w);
    ((float4*)s)[i] = make_float4(0.f, 0.f, 0.f, 0.f);
}

// ---------------------------------------------------------------------------
// SpMM (COO): one wave32 per edge; each lane handles 2 features.
// dst[row] += val * src[col]   (f32 atomics, L2-resident working set)
// ---------------------------------------------------------------------------
__global__ void ngcf_spmm(const float* __restrict__ vals,
                          const int*   __restrict__ rows,
                          const int*   __restrict__ cols,
                          const float* __restrict__ src,
                          float*       __restrict__ dst) {
    int edge = (blockIdx.x * blockDim.x + threadIdx.x) >> 5;
    int lane = threadIdx.x & 31;
    if (edge >= N_EDGES) return;
    int   col = cols[edge];
    int   row = rows[edge];
    float val = vals[edge];
    float2 sv = ((const float2*)(src + (size_t)col * EMB))[lane];
    float* d = dst + (size_t)row * EMB + lane * 2;
    atomicAdd(d + 0, val * sv.x);
    atomicAdd(d + 1, val * sv.y);
}

// ---------------------------------------------------------------------------
// Fused dense transform via V_WMMA_F32_16X16X4_F32 (pure fp32 path).
// One wave per 16-row tile: D[16x64] = S[16x64] @ W^T  (16 K-steps x 4 N-tiles)
// mode 0: e1 = S@W0^T; e2 = leaky(e1); out += 0.25*(e1+e2); e2out = e2; S tile := 0
// mode 1: e3 = S@W1^T; out += 0.25*e3
// ---------------------------------------------------------------------------
__global__ void ngcf_gemm_wmma(const float* __restrict__ S,
                               const float* __restrict__ W,
                               float* __restrict__ out,
                               float* __restrict__ e2out,
                               float* __restrict__ Szero,
                               int mode) {
    __shared__ float lW[EMB * EMB];                      // 16 KB: W staged once
    for (int i = threadIdx.x; i < EMB * EMB / 4; i += blockDim.x)
        ((float4*)lW)[i] = ((const float4*)W)[i];
    __syncthreads();

    const int wavesPerBlock = blockDim.x >> 5;
    int tile = blockIdx.x * wavesPerBlock + (threadIdx.x >> 5);
    if (tile >= N_NODES / 16) return;                    // wave-uniform guard

    const int lane  = threadIdx.x & 31;
    const int mrow  = lane & 15;                         // A: M = lane&15 / B,C,D: N = lane&15
    const int khalf = lane >> 4;                         // A: K-pair select / C,D: M-half select
    const int m0    = tile * 16;

    v8f acc[4] = {};                                     // 4 N-tiles of 16x16 f32
    const float* srow = S + (size_t)(m0 + mrow) * EMB + khalf * 2;

#pragma unroll
    for (int kk = 0; kk < 16; ++kk) {                    // K = 64 in steps of 4
        const int k0 = kk * 4;
        v2f a = *(const v2f*)(srow + k0);                // A 16x4 fragment (f32)
#pragma unroll
        for (int nt = 0; nt < 4; ++nt) {
            // B = W^T: B[k][n] = W[n][k]; lane n = nt*16 + mrow, k pair via khalf
            v2f b = *(const v2f*)(&lW[(nt * 16 + mrow) * EMB + k0 + khalf * 2]);
            acc[nt] = __builtin_amdgcn_wmma_f32_16x16x4_f32(
                false, a, false, b, (short)0, acc[nt], false, false);
        }
    }

    if (mode == 0) {
        // Re-zero this wave's S tile for the second SpMM (tile fully consumed above)
#pragma unroll
        for (int kk = 0; kk < 16; ++kk)
            *(float2*)(Szero + (size_t)(m0 + mrow) * EMB + kk * 4 + khalf * 2) =
                make_float2(0.f, 0.f);
    }

    // D layout: lane l, vgpr v -> m = v + 8*(l>>4), n = l&15
#pragma unroll
    for (int nt = 0; nt < 4; ++nt) {
#pragma unroll
        for (int v = 0; v < 8; ++v) {
            int    m   = m0 + v + 8 * khalf;
            int    n   = nt * 16 + mrow;
            size_t idx = (size_t)m * EMB + n;
            float  e1  = acc[nt][v];
            if (mode == 0) {
                float e2 = (e1 >= 0.f) ? e1 : LEAKY_ALPHA * e1;
                out[idx] += 0.25f * (e1 + e2);
                e2out[idx] = e2;
            } else {
                out[idx] += 0.25f * e1;
            }
        }
    }
}

// ---------------------------------------------------------------------------
extern "C" void kernel_launch(void* const* d_in, const int* in_sizes, int n_in,
                              void* d_out, int out_size, void* d_ws, size_t ws_size,
                              hipStream_t stream) {
    const float* user_emb = (const float*)d_in[0];
    const float* item_emb = (const float*)d_in[1];
    const float* W0       = (const float*)d_in[2];
    const float* W1       = (const float*)d_in[3];
    const float* vals     = (const float*)d_in[4];
    const int*   rows     = (const int*)d_in[5];
    const int*   cols     = (const int*)d_in[6];
    float* out = (float*)d_out;

    float* e0 = (float*)d_ws;                    // 9.6M floats (38.4 MB), later reused for e2
    float* s  = e0 + (size_t)N_NODES * EMB;      // 9.6M floats (38.4 MB) SpMM accumulator

    const int nElem4 = N_NODES * EMB / 4;
    ngcf_init<<<(nElem4 + 255) / 256, 256, 0, stream>>>(user_emb, item_emb, e0, s, out);

    const int spmmBlocks = (N_EDGES * 32 + 255) / 256;   // one wave32 per edge
    ngcf_spmm<<<spmmBlocks, 256, 0, stream>>>(vals, rows, cols, e0, s);

    const int tiles = N_NODES / 16;                      // 9375 tiles, 8 waves/block
    // e1/e2 stage: writes e2 into e0's buffer, zeroes s for the next SpMM
    ngcf_gemm_wmma<<<(tiles + 7) / 8, 256, 0, stream>>>(s, W0, out, e0, s, 0);

    ngcf_spmm<<<spmmBlocks, 256, 0, stream>>>(vals, rows, cols, e0, s);

    ngcf_gemm_wmma<<<(tiles + 7) / 8, 256, 0, stream>>>(s, W1, out, nullptr, nullptr, 1);
}